// RecurrentFullAttention_86071144612550
// MI455X (gfx1250) — compile-verified
//
#include <hip/hip_runtime.h>
#include <hip/hip_bf16.h>

// Problem constants (from reference): N=32, H=16, E=64, D=64, S=4096
#define NH    512          // N*H
#define SDIM  4096
#define SP1   4097
#define EDIM  64

typedef float v2f __attribute__((ext_vector_type(2)));
typedef float v4f __attribute__((ext_vector_type(4)));
typedef float v8f __attribute__((ext_vector_type(8)));

// --------------------------------------------------------------------------
// Kernel 1: per (n,h) compute scores q·k_s for s in [0,4096] into ws, while
// copying state_keys -> out_keys and appending `key` at s=4096.
//
// WMMA mapping (V_WMMA_F32_16X16X4_F32, chained over 16 K=4 chunks):
//   A[m,k] = q[e(c,k)] broadcast over all 16 rows
//   B[k,n] = keys[s0+n, e(c,k)]   (n = matrix column = s within tile)
// K-chunk -> e assignment chosen so each lane's operands are one contiguous
// 16B-aligned float4 per g=0..7:
//   chunk 2g  : (K0,K1)=e{8g,8g+1}   (half0) ; (K2,K3)=e{8g+4,8g+5} (half1)
//   chunk 2g+1: (K0,K1)=e{8g+2,8g+3} (half0) ; (K2,K3)=e{8g+6,8g+7} (half1)
// i.e. lane loads float4 at e = 8g + half*4; .xy feeds chunk 2g, .zw chunk 2g+1.
// All D rows identical => acc[0] in every lane = score for s0 + (lane&15).
// --------------------------------------------------------------------------
__global__ __launch_bounds__(256) void k1_scores_copy_keys(
    const float* __restrict__ q,        // [NH,64]
    const float* __restrict__ key,      // [NH,64]
    const float* __restrict__ sk,       // [NH,4096,64]
    float* __restrict__ out_keys,       // [NH,4097,64]
    float* __restrict__ scores)         // [NH,4097]
{
    const int nh   = blockIdx.x;
    const int yblk = blockIdx.y;            // 0 or 1: which half of the tiles
    const int lane = threadIdx.x & 31;
    const int wave = threadIdx.x >> 5;      // 0..7
    const int half = lane >> 4;             // 0/1: K-pair {0,1} vs {2,3}
    const int col  = lane & 15;             // matrix column (s within tile)

    const float* qrow = q + (size_t)nh * EDIM;

    // Broadcast-q A operands: float4 per g, .xy -> chunk 2g, .zw -> chunk 2g+1.
    v4f a4[8];
#pragma unroll
    for (int g = 0; g < 8; ++g)
        a4[g] = *(const v4f*)(qrow + g * 8 + half * 4);

    const float* skb = sk       + (size_t)nh * SDIM * EDIM;
    float*       okb = out_keys + (size_t)nh * SP1  * EDIM;
    float*       sb  = scores   + (size_t)nh * SP1;

    // 256 tiles of 16 s-rows split over 2 blocks; wave-uniform trip count
    // (EXEC all-ones across every WMMA).
    const int tile_end = (yblk + 1) * 128;
    for (int tile = yblk * 128 + wave; tile < tile_end; tile += 8) {
        const int s0 = tile * 16;
        const float* krow = skb + (size_t)(s0 + col) * EDIM;
        float*       orow = okb + (size_t)(s0 + col) * EDIM;

        // 8 independent 16B NT loads -> keep them grouped for MLP.
        v4f b4[8];
#pragma unroll
        for (int g = 0; g < 8; ++g)
            b4[g] = __builtin_nontemporal_load((const v4f*)(krow + g * 8 + half * 4));
        __builtin_amdgcn_sched_barrier(0);

        // Stream the cache copy from the same registers (zero extra reads).
#pragma unroll
        for (int g = 0; g < 8; ++g)
            __builtin_nontemporal_store(b4[g], (v4f*)(orow + g * 8 + half * 4));

        v8f acc = {};
#pragma unroll
        for (int g = 0; g < 8; ++g) {
            v2f alo = __builtin_shufflevector(a4[g], a4[g], 0, 1);
            v2f ahi = __builtin_shufflevector(a4[g], a4[g], 2, 3);
            v2f blo = __builtin_shufflevector(b4[g], b4[g], 0, 1);
            v2f bhi = __builtin_shufflevector(b4[g], b4[g], 2, 3);
            acc = __builtin_amdgcn_wmma_f32_16x16x4_f32(
                false, alo, false, blo, (short)0, acc, false, false);
            acc = __builtin_amdgcn_wmma_f32_16x16x4_f32(
                false, ahi, false, bhi, (short)0, acc, false, false);
        }

        // Every lane's acc[0] holds the score for s0+col (all rows identical).
        if (half == 0)
            sb[s0 + col] = acc[0];
    }

    // Appended key (s = 4096): one wave does a shuffle-reduced dot + copy.
    if (yblk == 0 && wave == 0) {
        const float* krow = key + (size_t)nh * EDIM;
        v2f kv = *(const v2f*)(krow + lane * 2);
        v2f qv = *(const v2f*)(qrow + lane * 2);
        float p = kv.x * qv.x + kv.y * qv.y;
#pragma unroll
        for (int off = 16; off > 0; off >>= 1)
            p += __shfl_xor(p, off, 32);
        __builtin_nontemporal_store(kv, (v2f*)(okb + (size_t)SDIM * EDIM + lane * 2));
        if (lane == 0)
            sb[SDIM] = p;
    }
}

// --------------------------------------------------------------------------
// Kernel 2: softmax over 4097 scores per (n,h), in place in ws.
// 256 threads; thread t owns indices t, t+256, ... (16 each); t==0 also 4096.
// --------------------------------------------------------------------------
__global__ __launch_bounds__(256) void k2_softmax(float* __restrict__ scores)
{
    const int nh   = blockIdx.x;
    const int t    = threadIdx.x;
    const int lane = t & 31;
    const int wave = t >> 5;
    float* s = scores + (size_t)nh * SP1;
    const float temp = 0.125f;   // 1/sqrt(64)

    float v[16];
    float m = -3.4e38f;
#pragma unroll
    for (int j = 0; j < 16; ++j) {
        v[j] = s[t + j * 256] * temp;
        m = fmaxf(m, v[j]);
    }
    float extra = 0.0f;
    if (t == 0) { extra = s[SDIM] * temp; m = fmaxf(m, extra); }

    __shared__ float redm[8];
    __shared__ float reds[8];

#pragma unroll
    for (int off = 16; off > 0; off >>= 1)
        m = fmaxf(m, __shfl_xor(m, off, 32));
    if (lane == 0) redm[wave] = m;
    __syncthreads();
    m = redm[0];
#pragma unroll
    for (int w = 1; w < 8; ++w) m = fmaxf(m, redm[w]);

    float sum = 0.0f;
#pragma unroll
    for (int j = 0; j < 16; ++j) {
        v[j] = __expf(v[j] - m);
        sum += v[j];
    }
    if (t == 0) { extra = __expf(extra - m); sum += extra; }

#pragma unroll
    for (int off = 16; off > 0; off >>= 1)
        sum += __shfl_xor(sum, off, 32);
    if (lane == 0) reds[wave] = sum;
    __syncthreads();
    sum = reds[0];
#pragma unroll
    for (int w = 1; w < 8; ++w) sum += reds[w];

    const float inv = 1.0f / sum;
#pragma unroll
    for (int j = 0; j < 16; ++j)
        s[t + j * 256] = v[j] * inv;
    if (t == 0)
        s[SDIM] = extra * inv;
}

// --------------------------------------------------------------------------
// Kernel 3: V[n,h,d] = sum_s a[s] * values[s,d], while copying
// state_values -> out_values and appending `value` at s=4096.
//
// Waves own ROW PAIRS (2 rows x 64 floats = 128 floats = 4/lane) so every
// lane does one aligned b128 NT load + store per pair. Lane half (lane>>4)
// selects row parity; lane&15 selects the d-quad. Unrolled x4 for MLP.
// Deterministic LDS reduction over 8 waves x 2 halves at the end.
// --------------------------------------------------------------------------
__global__ __launch_bounds__(256) void k3_values_out(
    const float* __restrict__ value,    // [NH,64]
    const float* __restrict__ sv,       // [NH,4096,64]
    const float* __restrict__ probs,    // [NH,4097]
    float* __restrict__ out_values,     // [NH,4097,64]
    float* __restrict__ out_V)          // [NH,64]
{
    const int nh   = blockIdx.x;
    const int lane = threadIdx.x & 31;
    const int wave = threadIdx.x >> 5;
    const int half = lane >> 4;         // row parity within the pair

    const float* svb = sv         + (size_t)nh * SDIM * EDIM;
    float*       ovb = out_values + (size_t)nh * SP1  * EDIM;
    const float* pa  = probs      + (size_t)nh * SP1;

    v4f acc = {0.0f, 0.0f, 0.0f, 0.0f};

    // 2048 row-pairs; wave w handles pairs w, w+8, ...; unroll 4 => 64 iters.
    for (int pb = wave; pb < SDIM / 2; pb += 32) {
        v4f  x[4];
        float av[4];
#pragma unroll
        for (int j = 0; j < 4; ++j) {
            const int p = pb + 8 * j;
            x[j]  = __builtin_nontemporal_load(
                        (const v4f*)(svb + (size_t)p * (2 * EDIM) + lane * 4));
            av[j] = pa[2 * p + half];
        }
        __builtin_amdgcn_sched_barrier(0);
#pragma unroll
        for (int j = 0; j < 4; ++j) {
            const int p = pb + 8 * j;
            __builtin_nontemporal_store(
                x[j], (v4f*)(ovb + (size_t)p * (2 * EDIM) + lane * 4));
            acc += x[j] * av[j];
        }
    }

    // Appended value row (s = 4096): wave 0, lanes 0-15 (one float4 each).
    if (wave == 0 && half == 0) {
        const float* vrow = value + (size_t)nh * EDIM;
        v4f x = *(const v4f*)(vrow + lane * 4);
        __builtin_nontemporal_store(
            x, (v4f*)(ovb + (size_t)SDIM * EDIM + lane * 4));
        acc += x * pa[SDIM];
    }

    __shared__ v4f red[8][32];
    red[wave][lane] = acc;
    __syncthreads();
    if (wave == 0 && half == 0) {
        v4f tot = red[0][lane] + red[0][lane + 16];
#pragma unroll
        for (int w = 1; w < 8; ++w)
            tot += red[w][lane] + red[w][lane + 16];
        *(v4f*)(out_V + (size_t)nh * EDIM + lane * 4) = tot;
    }
}

// --------------------------------------------------------------------------
extern "C" void kernel_launch(void* const* d_in, const int* in_sizes, int n_in,
                              void* d_out, int out_size, void* d_ws, size_t ws_size,
                              hipStream_t stream)
{
    const float* q   = (const float*)d_in[0];   // [32,16,64]
    const float* key = (const float*)d_in[1];   // [32,16,64]
    const float* val = (const float*)d_in[2];   // [32,16,64]
    const float* sk  = (const float*)d_in[3];   // [32,16,4096,64]
    const float* sv  = (const float*)d_in[4];   // [32,16,4096,64]

    float* out        = (float*)d_out;
    float* out_V      = out;                                   // 512*64
    float* out_keys   = out + (size_t)NH * EDIM;               // 512*4097*64
    float* out_values = out_keys + (size_t)NH * SP1 * EDIM;    // 512*4097*64

    float* scores = (float*)d_ws;                              // 512*4097 floats

    k1_scores_copy_keys<<<dim3(NH, 2), dim3(256), 0, stream>>>(q, key, sk, out_keys, scores);
    k2_softmax<<<dim3(NH), dim3(256), 0, stream>>>(scores);
    k3_values_out<<<dim3(NH), dim3(256), 0, stream>>>(val, sv, scores, out_values, out_V);
}